// SimpleGraphormer_81286551044268
// MI455X (gfx1250) — compile-verified
//
#include <hip/hip_runtime.h>
#include <hip/hip_bf16.h>

#define NN   50000
#define EE   800000
#define GG   128
#define DHW  128      // concat width per conv layer (4 heads * 32)
#define NHEAD 4
#define HIDW 32

typedef float v2f __attribute__((ext_vector_type(2)));
typedef float v8f __attribute__((ext_vector_type(8)));

// ---------------------------------------------------------------------------
// GEMM: out = in @ W^T + b for Q,K,V,Skip simultaneously.
// One 16x16 output tile per wave, K stepped by 4 via V_WMMA_F32_16X16X4_F32.
// N = 50000 = 3125*16 exactly -> no tail, EXEC stays all-ones for WMMA.
// A layout (16x4 f32): lane = 16*(k>=2)+m, K = vgpr + 2*(lane/16).
// B layout (4x16 f32): lane = 16*(k>=2)+n, K = vgpr + 2*(lane/16); B[k][n]=W[n][k].
// C/D (16x16 f32): vgpr r, lane l -> row r+8*(l/16), col l%16.
// ---------------------------------------------------------------------------
__global__ __launch_bounds__(128) void gemm_qkvs(
    const float* __restrict__ in, int Kdim,
    const float* __restrict__ Wq, const float* __restrict__ bq,
    const float* __restrict__ Wk, const float* __restrict__ bk,
    const float* __restrict__ Wv, const float* __restrict__ bv,
    const float* __restrict__ Ws, const float* __restrict__ bs,
    float* __restrict__ outQ, float* __restrict__ outK,
    float* __restrict__ outV, float* __restrict__ outS)
{
    const int wave = threadIdx.x >> 5;
    const int lane = threadIdx.x & 31;
    const int tile = blockIdx.x * 4 + wave;   // 0 .. 3125*32-1
    const int mt   = tile >> 5;               // row tile 0..3124
    const int nt   = tile & 31;               // 4 matrices x 8 col tiles
    const int mat  = nt >> 3;
    const int ct   = (nt & 7) * 16;           // col base within 128

    const float* W; const float* bias; float* out;
    if      (mat == 0) { W = Wq; bias = bq; out = outQ; }
    else if (mat == 1) { W = Wk; bias = bk; out = outK; }
    else if (mat == 2) { W = Wv; bias = bv; out = outV; }
    else               { W = Ws; bias = bs; out = outS; }

    const int row0 = mt * 16;
    const int lm = lane & 15;     // m (for A) / n (for B) / col (for C)
    const int lh = lane >> 4;     // k-group select

    const float* arow = in + (size_t)(row0 + lm) * Kdim + 2 * lh;
    const float* brow = W  + (size_t)(ct   + lm) * Kdim + 2 * lh;

    v8f c = {0.f,0.f,0.f,0.f,0.f,0.f,0.f,0.f};
    for (int k0 = 0; k0 < Kdim; k0 += 4) {
        v2f a = *(const v2f*)(arow + k0);
        v2f b = *(const v2f*)(brow + k0);
        c = __builtin_amdgcn_wmma_f32_16x16x4_f32(
                /*neg_a=*/false, a, /*neg_b=*/false, b,
                /*c_mod=*/(short)0, c, /*reuse_a=*/false, /*reuse_b=*/false);
    }
    const float bb = bias[ct + lm];
    float* orow = out + (size_t)(row0 + 8 * lh) * DHW + ct + lm;
#pragma unroll
    for (int r = 0; r < 8; ++r) orow[(size_t)r * DHW] = c[r] + bb;
}

// ---------------------------------------------------------------------------
// Edge-phase helpers
// ---------------------------------------------------------------------------
__device__ __forceinline__ unsigned f32_order_key(float f) {
    unsigned u = __float_as_uint(f);
    return (u & 0x80000000u) ? ~u : (u | 0x80000000u);
}
__device__ __forceinline__ float f32_order_dec(unsigned k) {
    return (k & 0x80000000u) ? __uint_as_float(k & 0x7fffffffu)
                             : __uint_as_float(~k);
}

__global__ void init_md(unsigned* __restrict__ mi, float* __restrict__ den, int n) {
    int t = blockIdx.x * blockDim.x + threadIdx.x;
    if (t < n) { mi[t] = 0u; den[t] = 0.f; }   // key 0 == -inf in ordered-uint space
}

// score[e,h] = <q[dst,h,:], k[src,h,:]> / sqrt(32); segment-max via atomicMax on keys
__global__ void edge_score(const int* __restrict__ src, const int* __restrict__ dst,
                           const float* __restrict__ q, const float* __restrict__ k,
                           float* __restrict__ score, unsigned* __restrict__ mi, int n)
{
    int t = blockIdx.x * blockDim.x + threadIdx.x;
    if (t >= n) return;
    int e = t >> 2, h = t & 3;
    int s = src[e], d = dst[e];
    const float4* qr = (const float4*)(q + (size_t)d * DHW + h * HIDW);
    const float4* kr = (const float4*)(k + (size_t)s * DHW + h * HIDW);
    float acc = 0.f;
#pragma unroll
    for (int i = 0; i < HIDW / 4; ++i) {
        float4 a = qr[i];
        float4 b = kr[i];
        acc = fmaf(a.x, b.x, acc);
        acc = fmaf(a.y, b.y, acc);
        acc = fmaf(a.z, b.z, acc);
        acc = fmaf(a.w, b.w, acc);
    }
    acc *= 0.17677669529663687f;   // 1/sqrt(32)
    score[t] = acc;
    atomicMax(mi + (size_t)d * 4 + h, f32_order_key(acc));
}

// ex = exp(score - max); den[dst,h] += ex   (score buffer overwritten with ex)
__global__ void edge_exp(const int* __restrict__ dst,
                         const unsigned* __restrict__ mi,
                         float* __restrict__ score, float* __restrict__ den, int n)
{
    int t = blockIdx.x * blockDim.x + threadIdx.x;
    if (t >= n) return;
    int e = t >> 2, h = t & 3;
    int d = dst[e];
    float m = f32_order_dec(mi[(size_t)d * 4 + h]);
    float ex = expf(score[t] - m);
    score[t] = ex;
    atomicAdd(den + (size_t)d * 4 + h, ex);
}

// hout[dst, dim..dim+3] += (ex/den) * v[src, dim..dim+3]
// one thread per (edge, 4-channel group): float4 gather of v, 4 atomic adds
__global__ void edge_agg(const int* __restrict__ src, const int* __restrict__ dst,
                         const float* __restrict__ ex, const float* __restrict__ den,
                         const float* __restrict__ v, float* __restrict__ hout, int n)
{
    int t = blockIdx.x * blockDim.x + threadIdx.x;
    if (t >= n) return;
    int e = t >> 5;                 // edge
    int g4 = t & 31;                // 4-channel group 0..31
    int dim = g4 * 4;
    int h = dim >> 5;               // head 0..3
    int s = src[e], d = dst[e];
    float alpha = ex[(size_t)e * 4 + h] / den[(size_t)d * 4 + h];
    float4 vv = *(const float4*)(v + (size_t)s * DHW + dim);
    float* o = hout + (size_t)d * DHW + dim;
    atomicAdd(o + 0, alpha * vv.x);
    atomicAdd(o + 1, alpha * vv.y);
    atomicAdd(o + 2, alpha * vv.z);
    atomicAdd(o + 3, alpha * vv.w);
}

// ---------------------------------------------------------------------------
// Pooling + MLP head
// ---------------------------------------------------------------------------
__global__ void init_pool(float* __restrict__ pooled, float* __restrict__ cnt, int n) {
    int t = blockIdx.x * blockDim.x + threadIdx.x;
    if (t < n) pooled[t] = 0.f;
    if (t < GG) cnt[t] = 0.f;
}

__global__ void pool_sum(const int* __restrict__ batch, const float* __restrict__ h,
                         float* __restrict__ pooled, float* __restrict__ cnt, int n)
{
    int t = blockIdx.x * blockDim.x + threadIdx.x;
    if (t >= n) return;
    int node = t >> 7, dim = t & 127;
    int g = batch[node];
    atomicAdd(pooled + (size_t)g * DHW + dim, h[t]);
    if (dim == 0) atomicAdd(cnt + g, 1.0f);
}

__global__ void mlp_head(const float* __restrict__ pooled, const float* __restrict__ cnt,
                         const float* __restrict__ w1, const float* __restrict__ b1,
                         const float* __restrict__ w2, const float* __restrict__ b2,
                         float* __restrict__ out)
{
    int g = threadIdx.x;
    if (g >= GG) return;
    float inv = 1.0f / fmaxf(cnt[g], 1.0f);
    const float* pr = pooled + (size_t)g * DHW;
    float z[HIDW];
#pragma unroll
    for (int j = 0; j < HIDW; ++j) {
        float s = b1[j];
        const float* wr = w1 + (size_t)j * DHW;
        for (int kk = 0; kk < DHW; ++kk) s = fmaf(pr[kk] * inv, wr[kk], s);
        z[j] = fmaxf(s, 0.f);
    }
    float o = b2[0];
#pragma unroll
    for (int j = 0; j < HIDW; ++j) o = fmaf(z[j], w2[j], o);
    out[g] = o;
}

// ---------------------------------------------------------------------------
extern "C" void kernel_launch(void* const* d_in, const int* in_sizes, int n_in,
                              void* d_out, int out_size, void* d_ws, size_t ws_size,
                              hipStream_t stream)
{
    const float* x     = (const float*)d_in[0];
    const int*   ei    = (const int*)d_in[1];
    const int*   srcp  = ei;
    const int*   dstp  = ei + EE;
    const int*   batch = (const int*)d_in[2];

    // params: per layer (jax sorted-key order): k_b,k_w,q_b,q_w,skip_b,skip_w,v_b,v_w
    struct Layer { const float *kb,*kw,*qb,*qw,*sb,*sw,*vb,*vw; };
    Layer L[3];
    int p = 3;
    for (int l = 0; l < 3; ++l) {
        L[l].kb = (const float*)d_in[p++]; L[l].kw = (const float*)d_in[p++];
        L[l].qb = (const float*)d_in[p++]; L[l].qw = (const float*)d_in[p++];
        L[l].sb = (const float*)d_in[p++]; L[l].sw = (const float*)d_in[p++];
        L[l].vb = (const float*)d_in[p++]; L[l].vw = (const float*)d_in[p++];
    }
    const float* b1 = (const float*)d_in[p++];
    const float* b2 = (const float*)d_in[p++];
    const float* w1 = (const float*)d_in[p++];
    const float* w2 = (const float*)d_in[p++];

    // Workspace carve-up (floats)
    float* ws = (float*)d_ws;
    size_t off = 0;
    float* qb   = ws + off; off += (size_t)NN * DHW;
    float* kb   = ws + off; off += (size_t)NN * DHW;
    float* vb   = ws + off; off += (size_t)NN * DHW;
    float* hA   = ws + off; off += (size_t)NN * DHW;
    float* hB   = ws + off; off += (size_t)NN * DHW;
    float* exb  = ws + off; off += (size_t)EE * NHEAD;
    unsigned* mi = (unsigned*)(ws + off); off += (size_t)NN * NHEAD;
    float* den  = ws + off; off += (size_t)NN * NHEAD;
    float* pooled = ws + off; off += (size_t)GG * DHW;
    float* cnt  = ws + off; off += (size_t)GG;

    const int GEMM_BLOCKS = (NN / 16) * 32 / 4;      // 3125*32 tiles, 4 waves/block
    const int E4  = EE * NHEAD;
    const int N4  = NN * NHEAD;
    const int EG4 = EE * 32;                         // (edge, 4-channel group) pairs
    const int NC  = NN * DHW;

    const float* hin = x;
    int Kdim = 16;
    float* hout = hA;
    for (int l = 0; l < 3; ++l) {
        gemm_qkvs<<<GEMM_BLOCKS, 128, 0, stream>>>(
            hin, Kdim,
            L[l].qw, L[l].qb, L[l].kw, L[l].kb,
            L[l].vw, L[l].vb, L[l].sw, L[l].sb,
            qb, kb, vb, hout);
        init_md<<<(N4 + 255) / 256, 256, 0, stream>>>(mi, den, N4);
        edge_score<<<(E4 + 255) / 256, 256, 0, stream>>>(srcp, dstp, qb, kb, exb, mi, E4);
        edge_exp<<<(E4 + 255) / 256, 256, 0, stream>>>(dstp, mi, exb, den, E4);
        edge_agg<<<(EG4 + 255) / 256, 256, 0, stream>>>(srcp, dstp, exb, den, vb, hout, EG4);

        hin  = hout;
        hout = (hout == hA) ? hB : hA;
        Kdim = DHW;
    }
    const float* hfinal = hin;   // output of layer 2

    init_pool<<<(GG * DHW + 255) / 256, 256, 0, stream>>>(pooled, cnt, GG * DHW);
    pool_sum<<<(NC + 255) / 256, 256, 0, stream>>>(batch, hfinal, pooled, cnt, NC);
    mlp_head<<<1, 128, 0, stream>>>(pooled, cnt, w1, b1, w2, b2, (float*)d_out);
}